// MLPSimConstructor_7172595384465
// MI455X (gfx1250) — compile-verified
//
#include <hip/hip_runtime.h>
#include <hip/hip_bf16.h>

// Problem constants from the reference
#define BB   32
#define NN   1024
#define TT   64
#define DD   32
#define MM   1088   // N + T, row/col stride of the output adjacency

typedef __attribute__((ext_vector_type(2))) float v2f;
typedef __attribute__((ext_vector_type(8))) float v8f;

// tanh(relu(x)). gfx1250 has a V_TANH_F32 transcendental; fall back to
// exp + native rcp (NOT full IEEE division) if the builtin is absent.
__device__ __forceinline__ float tanh_relu(float x) {
    float v = fmaxf(x, 0.0f);
#if __has_builtin(__builtin_amdgcn_tanhf)
    return __builtin_amdgcn_tanhf(v);
#else
    float e = __expf(-2.0f * v);
    return (1.0f - e) * __builtin_amdgcn_rcpf(1.0f + e);
#endif
}

__device__ __forceinline__ float fast_sigmoid(float z) {
    return __builtin_amdgcn_rcpf(1.0f + __expf(-z));
}

__device__ __forceinline__ v8f wmma_f32(v2f a, v2f b, v8f c) {
    return __builtin_amdgcn_wmma_f32_16x16x4_f32(
        /*neg_a=*/false, a, /*neg_b=*/false, b,
        /*c_mod=*/(short)0, c, /*reuse_a=*/false, /*reuse_b=*/false);
}

// ---------------------------------------------------------------------------
// Gram diagonal block: out[b, off+i, off+j] = tanh_relu(dot(x[b,i,:], x[b,j,:]))
// One wave computes a 2x2 group of 16x16 tiles (32x32 output) via 32 chained
// V_WMMA_F32_16X16X4_F32; each A/B fragment is loaded once, used twice.
// gridDim.x = (L/32)^2 / (blockDim.x/32); gridDim.y = B.
// ---------------------------------------------------------------------------
__global__ void gram_diag_kernel(const float* __restrict__ x, float* __restrict__ out,
                                 int L, int diag_off) {
    const int wave = threadIdx.x >> 5;
    const int lane = threadIdx.x & 31;
    const int gpr  = L >> 5;                          // 32-wide groups per row
    const int g    = blockIdx.x * (blockDim.x >> 5) + wave;
    const int gi   = g / gpr;
    const int gj   = g % gpr;
    const int b    = blockIdx.y;

    const float* xb = x + (size_t)b * L * DD;

    const int hi   = lane >> 4;   // lane-half selects K pair (and row group for C)
    const int mrow = lane & 15;

    // Fragment layout (16x4 f32): lane holds row (l&15), float2 at col 4*kc + 2*(l>>4)
    const float* a0 = xb + (size_t)(gi * 32 + mrow) * DD + 2 * hi;
    const float* a1 = a0 + 16 * DD;
    const float* b0 = xb + (size_t)(gj * 32 + mrow) * DD + 2 * hi;
    const float* b1 = b0 + 16 * DD;

    v8f acc00 = {}, acc01 = {}, acc10 = {}, acc11 = {};
#pragma unroll
    for (int kc = 0; kc < 8; ++kc) {
        v2f af0 = *(const v2f*)(a0 + 4 * kc);
        v2f af1 = *(const v2f*)(a1 + 4 * kc);
        v2f bf0 = *(const v2f*)(b0 + 4 * kc);
        v2f bf1 = *(const v2f*)(b1 + 4 * kc);
        acc00 = wmma_f32(af0, bf0, acc00);
        acc01 = wmma_f32(af0, bf1, acc01);
        acc10 = wmma_f32(af1, bf0, acc10);
        acc11 = wmma_f32(af1, bf1, acc11);
    }

    // C layout: VGPR r -> row (r + 8*hi), col (lane&15)
    const int ncol  = lane & 15;
    const int mbase = hi * 8;
    size_t base = (size_t)b * MM * MM
                + (size_t)(diag_off + gi * 32 + mbase) * MM
                + (size_t)(diag_off + gj * 32 + ncol);
#pragma unroll
    for (int r = 0; r < 8; ++r) {
        out[base + (size_t)r * MM]             = tanh_relu(acc00[r]);
        out[base + (size_t)r * MM + 16]        = tanh_relu(acc01[r]);
        out[base + (size_t)(16 + r) * MM]      = tanh_relu(acc10[r]);
        out[base + (size_t)(16 + r) * MM + 16] = tanh_relu(acc11[r]);
    }
}

// ---------------------------------------------------------------------------
// Cross block (ST / TS): out[b, row_off+r, col_off+c] =
//   tanh_relu( b2 + sum_d sigmoid(A[r,d] + Bt[c,d]) * w2[d] )
// where A = x1_tile @ w1[:32], Bt = x2_tile @ w1[32:] + b1 (computed in-block).
// Block handles a 16-row x 64-col tile with 256 threads (4 outputs/thread).
// grid = (Rtotal/16, Ctotal/64, B)
// ---------------------------------------------------------------------------
__global__ void cross_block_kernel(const float* __restrict__ x1, const float* __restrict__ x2,
                                   const float* __restrict__ w1, const float* __restrict__ b1,
                                   const float* __restrict__ w2, const float* __restrict__ b2,
                                   float* __restrict__ out,
                                   int Rtotal, int Ctotal, int row_off, int col_off) {
    const int b  = blockIdx.z;
    const int r0 = blockIdx.x * 16;
    const int c0 = blockIdx.y * 64;
    const int t  = threadIdx.x;   // 0..255

    __shared__ float A[16][DD];
    __shared__ float Bt[64][DD + 1];   // pad: lane-indexed rows -> conflict-free
    __shared__ float W2[DD];

    const float* x1b = x1 + ((size_t)b * Rtotal + r0) * DD;
    const float* x2b = x2 + ((size_t)b * Ctotal + c0) * DD;

    // A tile: 16x32 elems, first-layer matmul (row half of w1, no bias)
    for (int i = t; i < 16 * DD; i += 256) {
        int rr = i >> 5, d = i & 31;
        float acc = 0.0f;
#pragma unroll
        for (int k = 0; k < DD; ++k) acc += x1b[rr * DD + k] * w1[k * DD + d];
        A[rr][d] = acc;
    }
    // Bt tile: 64x32 elems (col half of w1, + b1 folded in)
    for (int i = t; i < 64 * DD; i += 256) {
        int cc = i >> 5, d = i & 31;
        float acc = b1[d];
#pragma unroll
        for (int k = 0; k < DD; ++k) acc += x2b[cc * DD + k] * w1[(DD + k) * DD + d];
        Bt[cc][d] = acc;
    }
    if (t < DD) W2[t] = w2[t];
    __syncthreads();

    const float bias2 = b2[0];
    const int c = t & 63;        // column within tile (lane-contiguous)
    const int rbase = t >> 6;    // 0..3
#pragma unroll
    for (int s = 0; s < 4; ++s) {
        int r = rbase * 4 + s;
        float acc = bias2;
#pragma unroll
        for (int d = 0; d < DD; ++d) {
            acc += fast_sigmoid(A[r][d] + Bt[c][d]) * W2[d];
        }
        out[(size_t)b * MM * MM + (size_t)(row_off + r0 + r) * MM + (col_off + c0 + c)]
            = tanh_relu(acc);
    }
}

extern "C" void kernel_launch(void* const* d_in, const int* in_sizes, int n_in,
                              void* d_out, int out_size, void* d_ws, size_t ws_size,
                              hipStream_t stream) {
    const float* spatial  = (const float*)d_in[0];  // [B, N, D]
    const float* temporal = (const float*)d_in[1];  // [B, T, D]
    const float* st_w1 = (const float*)d_in[2];     // [2D, D]
    const float* st_b1 = (const float*)d_in[3];     // [D]
    const float* st_w2 = (const float*)d_in[4];     // [D, 1]
    const float* st_b2 = (const float*)d_in[5];     // [1]
    const float* ts_w1 = (const float*)d_in[6];
    const float* ts_b1 = (const float*)d_in[7];
    const float* ts_w2 = (const float*)d_in[8];
    const float* ts_b2 = (const float*)d_in[9];
    float* out = (float*)d_out;                     // [B, 1088, 1088]

    // Spatial x Spatial Gram (1024x1024): (32)^2 = 1024 groups / 8 waves
    gram_diag_kernel<<<dim3(128, BB), dim3(256), 0, stream>>>(spatial, out, NN, 0);
    // Temporal x Temporal Gram (64x64): 4 groups, 4-wave block
    gram_diag_kernel<<<dim3(1, BB), dim3(128), 0, stream>>>(temporal, out, TT, NN);

    // ST block: rows = spatial (N), cols = temporal (T) -> out[:, :N, N:]
    cross_block_kernel<<<dim3(NN / 16, TT / 64, BB), dim3(256), 0, stream>>>(
        spatial, temporal, st_w1, st_b1, st_w2, st_b2, out, NN, TT, 0, NN);
    // TS block: rows = temporal (T), cols = spatial (N) -> out[:, N:, :N]
    cross_block_kernel<<<dim3(TT / 16, NN / 64, BB), dim3(256), 0, stream>>>(
        temporal, spatial, ts_w1, ts_b1, ts_w2, ts_b2, out, TT, NN, NN, 0);
}